// MambaHeadWithWrapper_37641093382896
// MI455X (gfx1250) — compile-verified
//
#include <hip/hip_runtime.h>

// ---------------------------------------------------------------------------
// Fused Mamba (4 layers) + head for B=32768, L=10, D=8, ED=16, N=16, DCONV=4.
// One wave32 processes 16 sequences end-to-end.  All projections (in_proj,
// x_proj Bm/Cm/dt, out_proj) run on V_WMMA_F32_16X16X4_F32 (exact fp32).
// The dt row is computed by a broadcast-column WMMA (B operand = x_proj row 0
// replicated across all 16 columns), so every lane receives dt[b] directly in
// its accumulator -- no serialized ds_bpermute reduction chain.
// Selective-scan state h[b][e][n] lives in registers (128 VGPRs/lane).
// LDS staging is wave-private (same-wave DS ops are in-order).
// Transcendentals are raw v_exp_f32 / v_log_f32 / v_rsq_f32 / v_rcp_f32 with
// log2(e) folded into the preloaded A matrix.
// ---------------------------------------------------------------------------

#define SEQ_L   10
#define DIM_D   8
#define NLAYER  4
#define EDIM    16
#define NSTATE  16

#define LOG2E_F 1.4426950408889634f
#define LN2_F   0.6931471805599453f

typedef float v2f __attribute__((ext_vector_type(2)));
typedef float v8f __attribute__((ext_vector_type(8)));

struct MambaParams {
  const float* x;
  const float* A_log[NLAYER];
  const float* Dp[NLAYER];
  const float* conv_b[NLAYER];
  const float* conv_w[NLAYER];
  const float* dt_proj_b[NLAYER];
  const float* dt_proj_w[NLAYER];
  const float* in_proj[NLAYER];
  const float* norm_w[NLAYER];
  const float* out_proj[NLAYER];
  const float* x_proj[NLAYER];
  const float* norm_f;
  const float* fc_w;
  const float* fc_b;
  float* out;
};

__device__ __forceinline__ v8f wmma4(v2f a, v2f b, v8f c) {
  // D = A(16x4,f32) * B(4x16,f32) + C(16x16,f32)
  return __builtin_amdgcn_wmma_f32_16x16x4_f32(
      /*neg_a=*/false, a, /*neg_b=*/false, b,
      /*c_mod=*/(short)0, c, /*reuse_a=*/false, /*reuse_b=*/false);
}

// Raw hardware transcendentals (v_exp_f32 is 2^x, v_log_f32 is log2(x)).
__device__ __forceinline__ float exp2_hw(float x) { return __builtin_amdgcn_exp2f(x); }
__device__ __forceinline__ float log2_hw(float x) { return __builtin_amdgcn_logf(x); }
__device__ __forceinline__ float rcp_hw(float x)  { return __builtin_amdgcn_rcpf(x); }
__device__ __forceinline__ float rsq_hw(float x)  { return __builtin_amdgcn_rsqf(x); }

__device__ __forceinline__ float softplus_f(float x) {
  // log(1+e^x) = ln2 * log2(1 + 2^(x*log2e)); x is small here, guard anyway.
  if (x > 20.0f) return x;
  return LN2_F * log2_hw(1.0f + exp2_hw(x * LOG2E_F));
}
__device__ __forceinline__ float silu_f(float x) {
  // x * sigmoid(x) with v_rcp_f32 (no IEEE divide sequence)
  return x * rcp_hw(1.0f + exp2_hw(-x * LOG2E_F));
}
__device__ __forceinline__ void wave_sync() {
  __builtin_amdgcn_wave_barrier();   // same-wave LDS is in-order in HW
}

extern "C" __global__ __launch_bounds__(256, 1)
void mamba_fused_kernel(MambaParams P) {
  const int lane  = threadIdx.x & 31;
  const int wv    = threadIdx.x >> 5;    // wave within block (0..7)
  const int laneN = lane & 15;           // N/M row index inside a 16x16 tile
  const int hi    = lane >> 4;           // lane-half (selects K/M offsets)
  const int b0    = blockIdx.x * 128 + wv * 16;  // first sequence of this wave
  const int bN    = b0 + laneN;

  __shared__ float xst [8][256];   // xc tile staged as [b][e] for A-layout reads
  __shared__ float bmst[8][256];   // Bm tile [b][n]
  __shared__ float cmst[8][256];   // Cm tile [b][n]
  __shared__ float yst [8][256];   // gated y tile [b][e]
  __shared__ float ost [8][128];   // out_proj tile [b][d]

  // Lane's d-slots inside the A-layout: hi=0 -> {0,1,4,5}, hi=1 -> {2,3,6,7}
  int dmapv[4];
#pragma unroll
  for (int j = 0; j < 4; ++j) dmapv[j] = ((j >> 1) * 4) + (j & 1) + 2 * hi;

  // ---- residual stream in WMMA-A layout: res[l][j] = h[bN][l][dmapv[j]] ----
  float res[SEQ_L][4];
#pragma unroll
  for (int l = 0; l < SEQ_L; ++l)
#pragma unroll
    for (int j = 0; j < 4; ++j)
      res[l][j] = P.x[bN * (SEQ_L * DIM_D) + l * DIM_D + dmapv[j]];

  for (int li = 0; li < NLAYER; ++li) {
    const float* Alog = P.A_log[li];
    const float* Dpar = P.Dp[li];
    const float* cb_p = P.conv_b[li];
    const float* cw_p = P.conv_w[li];
    const float* dtb_p = P.dt_proj_b[li];
    const float* dtw_p = P.dt_proj_w[li];
    const float* Wip  = P.in_proj[li];    // (32,8) row-major
    const float* nrm  = P.norm_w[li];     // (8,)
    const float* Wout = P.out_proj[li];   // (8,16) row-major
    const float* Xp   = P.x_proj[li];     // (33,16) row-major

    // ---- small per-layer params into registers -------------------------
    float nw[4];
#pragma unroll
    for (int j = 0; j < 4; ++j) nw[j] = nrm[dmapv[j]];

    float wip[2][2][2];                  // [tile(xl,z)][kchunk][j2]
#pragma unroll
    for (int t = 0; t < 2; ++t)
#pragma unroll
      for (int ki = 0; ki < 2; ++ki)
#pragma unroll
        for (int j2 = 0; j2 < 2; ++j2)
          wip[t][ki][j2] = Wip[(t * 16 + laneN) * 8 + ki * 4 + j2 + 2 * hi];

    float cw[4];
#pragma unroll
    for (int k = 0; k < 4; ++k) cw[k] = cw_p[laneN * 4 + k];
    const float cb  = cb_p[laneN];
    const float dtw = dtw_p[laneN];
    const float dtb = dtb_p[laneN];
    const float Dsk = Dpar[laneN];

    // x_proj B-operands: Bm rows (f=1..16), Cm rows (f=17..32), and the dt
    // row (f=0) broadcast across all 16 output columns.
    float wxB[4][2], wxC[4][2], wxD[4][2];
#pragma unroll
    for (int ki = 0; ki < 4; ++ki)
#pragma unroll
      for (int j2 = 0; j2 < 2; ++j2) {
        wxB[ki][j2] = Xp[(1  + laneN) * 16 + ki * 4 + j2 + 2 * hi];
        wxC[ki][j2] = Xp[(17 + laneN) * 16 + ki * 4 + j2 + 2 * hi];
        wxD[ki][j2] = Xp[/*f=0*/          ki * 4 + j2 + 2 * hi];   // same for all lanes
      }

    float wop[4][2];                           // out_proj B-operand (cols d<8)
#pragma unroll
    for (int ki = 0; ki < 4; ++ki)
#pragma unroll
      for (int j2 = 0; j2 < 2; ++j2)
        wop[ki][j2] = (laneN < 8) ? Wout[laneN * 16 + ki * 4 + j2 + 2 * hi] : 0.0f;

    // A[e=laneN][n] = -exp(A_log), with log2(e) pre-folded so the scan's
    // deltaA = 2^(delta * An[n]) is a single v_mul + v_exp_f32.
    float An[NSTATE];
#pragma unroll
    for (int n = 0; n < NSTATE; ++n)
      An[n] = -exp2_hw(Alog[laneN * 16 + n] * LOG2E_F) * LOG2E_F;

    // ---- per-timestep state --------------------------------------------
    float xh0[8], xh1[8], xh2[8];              // conv history (l-1, l-2, l-3)
    float h[8][NSTATE];                        // scan state h[b=v+8hi][e=laneN][n]
#pragma unroll
    for (int v = 0; v < 8; ++v) {
      xh0[v] = xh1[v] = xh2[v] = 0.0f;
#pragma unroll
      for (int n = 0; n < NSTATE; ++n) h[v][n] = 0.0f;
    }

#pragma unroll
    for (int l = 0; l < SEQ_L; ++l) {
      // -------- rmsnorm (cross-half reduction over d) -------------------
      float ss = res[l][0]*res[l][0] + res[l][1]*res[l][1]
               + res[l][2]*res[l][2] + res[l][3]*res[l][3];
      ss += __shfl_xor(ss, 16, 32);
      const float rstd = rsq_hw(ss * 0.125f + 1e-5f);
      float nx[4];
#pragma unroll
      for (int j = 0; j < 4; ++j) nx[j] = res[l][j] * rstd * nw[j];

      // -------- in_proj : 2 N-tiles x 2 K-chunks of WMMA ---------------
      v2f a_k0 = { nx[0], nx[1] };   // K 0..3
      v2f a_k4 = { nx[2], nx[3] };   // K 4..7
      v8f accx = { 0,0,0,0,0,0,0,0 };
      v8f accz = { 0,0,0,0,0,0,0,0 };
      accx = wmma4(a_k0, (v2f){ wip[0][0][0], wip[0][0][1] }, accx);
      accx = wmma4(a_k4, (v2f){ wip[0][1][0], wip[0][1][1] }, accx);
      accz = wmma4(a_k0, (v2f){ wip[1][0][0], wip[1][0][1] }, accz);
      accz = wmma4(a_k4, (v2f){ wip[1][1][0], wip[1][1][1] }, accz);

      // -------- depthwise conv (k=3 is current step) + SiLU ------------
      float xc[8], zv[8];
#pragma unroll
      for (int v = 0; v < 8; ++v) {
        const float xl_ = accx[v];
        const float c = cw[3]*xl_ + cw[2]*xh0[v] + cw[1]*xh1[v] + cw[0]*xh2[v] + cb;
        xc[v] = silu_f(c);
        zv[v] = accz[v];
        xh2[v] = xh1[v]; xh1[v] = xh0[v]; xh0[v] = xl_;
      }

      // -------- stage xc [b][e]; x_proj (Bm, Cm, dt) via WMMA ----------
#pragma unroll
      for (int v = 0; v < 8; ++v) xst[wv][(v + 8*hi) * 16 + laneN] = xc[v];
      wave_sync();

      v8f bm  = { 0,0,0,0,0,0,0,0 };
      v8f cm  = { 0,0,0,0,0,0,0,0 };
      v8f dtt = { 0,0,0,0,0,0,0,0 };   // every column = dt[b] (broadcast B)
#pragma unroll
      for (int ki = 0; ki < 4; ++ki) {
        v2f a = { xst[wv][laneN * 16 + ki * 4 + 0 + 2 * hi],
                  xst[wv][laneN * 16 + ki * 4 + 1 + 2 * hi] };
        bm  = wmma4(a, (v2f){ wxB[ki][0], wxB[ki][1] }, bm);
        cm  = wmma4(a, (v2f){ wxC[ki][0], wxC[ki][1] }, cm);
        dtt = wmma4(a, (v2f){ wxD[ki][0], wxD[ki][1] }, dtt);
      }
#pragma unroll
      for (int v = 0; v < 8; ++v) {
        bmst[wv][(v + 8*hi) * 16 + laneN] = bm[v];
        cmst[wv][(v + 8*hi) * 16 + laneN] = cm[v];
      }
      wave_sync();

      // -------- delta = softplus(dt[b]*dtw[e] + dtb[e]) ----------------
      float delta[8];
#pragma unroll
      for (int v = 0; v < 8; ++v)
        delta[v] = softplus_f(dtt[v] * dtw + dtb);

      // -------- selective scan update + C contraction ------------------
      float yv[8];
#pragma unroll
      for (int v = 0; v < 8; ++v) {
        const float u = delta[v] * xc[v];
        const float4* bmrow = (const float4*)&bmst[wv][(v + 8*hi) * 16];
        const float4* cmrow = (const float4*)&cmst[wv][(v + 8*hi) * 16];
        float acc = 0.0f;
#pragma unroll
        for (int q = 0; q < 4; ++q) {
          const float4 bq = bmrow[q];
          const float4 cq = cmrow[q];
          const float bqa[4] = { bq.x, bq.y, bq.z, bq.w };
          const float cqa[4] = { cq.x, cq.y, cq.z, cq.w };
#pragma unroll
          for (int r = 0; r < 4; ++r) {
            const int n = q * 4 + r;
            const float dA = exp2_hw(delta[v] * An[n]);   // log2e folded in An
            h[v][n] = dA * h[v][n] + u * bqa[r];
            acc += h[v][n] * cqa[r];
          }
        }
        yv[v] = acc + Dsk * xc[v];
      }

      // -------- gate with silu(z), stage, out_proj via WMMA ------------
#pragma unroll
      for (int v = 0; v < 8; ++v) {
        yv[v] *= silu_f(zv[v]);
        yst[wv][(v + 8*hi) * 16 + laneN] = yv[v];
      }
      wave_sync();

      v8f oacc = { 0,0,0,0,0,0,0,0 };
#pragma unroll
      for (int ki = 0; ki < 4; ++ki) {
        v2f a = { yst[wv][laneN * 16 + ki * 4 + 0 + 2 * hi],
                  yst[wv][laneN * 16 + ki * 4 + 1 + 2 * hi] };
        oacc = wmma4(a, (v2f){ wop[ki][0], wop[ki][1] }, oacc);
      }
      if (laneN < 8) {
#pragma unroll
        for (int v = 0; v < 8; ++v)
          ost[wv][(v + 8*hi) * 8 + laneN] = oacc[v];
      }
      wave_sync();

      // -------- residual add back in A-layout --------------------------
#pragma unroll
      for (int j = 0; j < 4; ++j)
        res[l][j] += ost[wv][laneN * 8 + dmapv[j]];
    } // timestep l
  }   // layer

  // ---- final rmsnorm + sliced FC head + ReLU -----------------------------
  float fw[7];
#pragma unroll
  for (int s = 0; s < 7; ++s) fw[s] = P.fc_w[s];
  const float fb = P.fc_b[0];
  float nf[4];
#pragma unroll
  for (int j = 0; j < 4; ++j) nf[j] = P.norm_f[dmapv[j]];

  float acc[4] = { 0.0f, 0.0f, 0.0f, 0.0f };
#pragma unroll
  for (int s = 0; s < 7; ++s) {
    const int l = (SEQ_L - 7) + s;  // last 7 timesteps
    float ss = res[l][0]*res[l][0] + res[l][1]*res[l][1]
             + res[l][2]*res[l][2] + res[l][3]*res[l][3];
    ss += __shfl_xor(ss, 16, 32);
    const float rstd = rsq_hw(ss * 0.125f + 1e-5f);
#pragma unroll
    for (int j = 0; j < 4; ++j)
      acc[j] += res[l][j] * rstd * nf[j] * fw[s];
  }
#pragma unroll
  for (int j = 0; j < 4; ++j) {
    const int c = dmapv[j];
    if (c < 6) {                     // first LBL_LEN*LBL_DIM=6 channels only
      const float o = acc[j] + fb;
      P.out[bN * 6 + c] = (o > 0.0f) ? o : 0.0f;
    }
  }
}

// ---------------------------------------------------------------------------
// Host-side launch.  Input ordering assumes jax pytree flattening of the
// setup_inputs() dict: x, params{layers[4]{sorted keys}, norm_f}, fc_w, fc_b.
// Per-layer sorted keys: A_log, D, conv_b, conv_w, dt_proj_b, dt_proj_w,
//                        in_proj, norm_w, out_proj, x_proj   (10 leaves)
// => d_in[0]=x, d_in[1+10*i+k]=layer i leaf k, d_in[41]=norm_f,
//    d_in[42]=fc_w, d_in[43]=fc_b.
// ---------------------------------------------------------------------------
extern "C" void kernel_launch(void* const* d_in, const int* in_sizes, int n_in,
                              void* d_out, int out_size, void* d_ws, size_t ws_size,
                              hipStream_t stream) {
  (void)in_sizes; (void)n_in; (void)out_size; (void)d_ws; (void)ws_size;

  MambaParams P;
  P.x = (const float*)d_in[0];
  for (int i = 0; i < NLAYER; ++i) {
    const int base = 1 + i * 10;
    P.A_log[i]     = (const float*)d_in[base + 0];
    P.Dp[i]        = (const float*)d_in[base + 1];
    P.conv_b[i]    = (const float*)d_in[base + 2];
    P.conv_w[i]    = (const float*)d_in[base + 3];
    P.dt_proj_b[i] = (const float*)d_in[base + 4];
    P.dt_proj_w[i] = (const float*)d_in[base + 5];
    P.in_proj[i]   = (const float*)d_in[base + 6];
    P.norm_w[i]    = (const float*)d_in[base + 7];
    P.out_proj[i]  = (const float*)d_in[base + 8];
    P.x_proj[i]    = (const float*)d_in[base + 9];
  }
  P.norm_f = (const float*)d_in[41];
  P.fc_w   = (const float*)d_in[42];
  P.fc_b   = (const float*)d_in[43];
  P.out    = (float*)d_out;

  // 32768 sequences / (8 waves * 16 seq per wave) = 256 blocks of 256 threads
  mamba_fused_kernel<<<dim3(256), dim3(256), 0, stream>>>(P);
}